// MeanAggregator_62775241999125
// MI455X (gfx1250) — compile-verified
//
#include <hip/hip_runtime.h>

// Problem constants (from the reference)
#define NROWS 50000
#define KNBR  32
#define DIN   256
#define DCOMB 512   // 2 * DIN
#define DOUT  256

#define BLOCK_ROWS 32
#define ROW_STRIDE 516   // 512 + 4 floats pad -> 4-bank rotation per row (64 banks)

typedef __attribute__((ext_vector_type(2))) float v2f;
typedef __attribute__((ext_vector_type(4))) float v4f;
typedef __attribute__((ext_vector_type(8))) float v8f;

__launch_bounds__(256)
__global__ void sage_fused_wmma_kernel(const float* __restrict__ node,
                                       const float* __restrict__ nbr,
                                       const float* __restrict__ W,
                                       const float* __restrict__ bias,
                                       float* __restrict__ out)
{
    __shared__ float smem[BLOCK_ROWS * ROW_STRIDE];

    const int tid  = threadIdx.x;
    const int row0 = blockIdx.x * BLOCK_ROWS;

    // ---------------- Phase 1: mean-aggregate + concat into LDS ----------------
    // Thread (tid) owns float4 column c4 of rows rg, rg+4, ..., rg+28.
    // Row index is CLAMPED (NROWS % 16 == 0): in the one tail block the
    // invalid row-tile reads duplicates of row NROWS-1 (valid addresses) and
    // its results are never stored. Fully branchless, single code copy.
    const int c4 = (tid & 63) << 2;   // 0..252, step 4
    const int rg = tid >> 6;          // 0..3

    #pragma unroll
    for (int rr = rg; rr < BLOCK_ROWS; rr += 4) {
        const int row = min(row0 + rr, NROWS - 1);
        const v4f self = *(const v4f*)&node[(size_t)row * DIN + c4];
        const float* p = &nbr[(size_t)row * (KNBR * DIN) + c4];
        v4f acc = {0.0f, 0.0f, 0.0f, 0.0f};
        #pragma unroll 8
        for (int k = 0; k < KNBR; ++k) {
            acc += *(const v4f*)(p + (size_t)k * DIN);
        }
        acc *= (1.0f / (float)KNBR);
        *(v4f*)&smem[rr * ROW_STRIDE + c4]       = self;   // combined[:, 0:256]
        *(v4f*)&smem[rr * ROW_STRIDE + DIN + c4] = acc;    // combined[:, 256:512]
    }
    __syncthreads();

    // ---------------- Phase 2: [32 x 512] @ [512 x 256]^T via f32 WMMA ----------------
    // Wave w: row tile (w & 1), column tiles (w>>1) + {0,4,8,12}.
    const int lane  = tid & 31;
    const int wave  = tid >> 5;
    const int mn    = lane & 15;             // M for A / N for B
    const int khalf = (lane >> 4) << 1;      // lanes 0-15 -> K0,K1 ; lanes 16-31 -> K2,K3

    const int rowtile = wave & 1;
    const int cbase   = wave >> 1;           // 0..3

    const float* arow = &smem[(rowtile * 16 + mn) * ROW_STRIDE + khalf];
    const float* b0 = &W[(size_t)((cbase * 16 +   0) + mn) * DCOMB + khalf];
    const float* b1 = &W[(size_t)((cbase * 16 +  64) + mn) * DCOMB + khalf];
    const float* b2 = &W[(size_t)((cbase * 16 + 128) + mn) * DCOMB + khalf];
    const float* b3 = &W[(size_t)((cbase * 16 + 192) + mn) * DCOMB + khalf];

    v8f acc0 = {}, acc1 = {}, acc2 = {}, acc3 = {};

    #pragma unroll 4
    for (int ks = 0; ks < DCOMB; ks += 4) {
        const v2f a  = *(const v2f*)(arow + ks);     // ds_load_b64 (paired by compiler), reused 4x
        const v2f w0 = *(const v2f*)(b0 + ks);       // global_load_b64 (L2-resident W)
        const v2f w1 = *(const v2f*)(b1 + ks);
        const v2f w2 = *(const v2f*)(b2 + ks);
        const v2f w3 = *(const v2f*)(b3 + ks);
        acc0 = __builtin_amdgcn_wmma_f32_16x16x4_f32(false, a, false, w0, (short)0, acc0, false, false);
        acc1 = __builtin_amdgcn_wmma_f32_16x16x4_f32(false, a, false, w1, (short)0, acc1, false, false);
        acc2 = __builtin_amdgcn_wmma_f32_16x16x4_f32(false, a, false, w2, (short)0, acc2, false, false);
        acc3 = __builtin_amdgcn_wmma_f32_16x16x4_f32(false, a, false, w3, (short)0, acc3, false, false);
    }

    // ---------------- Epilogue: bias + ReLU + store ----------------
    // C/D layout (16x16 f32): VGPR v holds M = v + 8*(lane>=16), N = lane%16.
    // Since NROWS % 16 == 0, validity is per ROW-TILE -> wave-uniform scalar
    // branch around fully unguarded stores (no per-lane exec masking).
    const int tilerow0 = row0 + rowtile * 16;
    if (tilerow0 + 16 <= NROWS) {
        const int mbase = tilerow0 + ((lane >> 4) << 3);
        float* const obase = out + (size_t)mbase * DOUT;
        #pragma unroll
        for (int j = 0; j < 4; ++j) {
            const v8f acc = (j == 0) ? acc0 : (j == 1) ? acc1 : (j == 2) ? acc2 : acc3;
            const int col = (cbase * 16 + 64 * j) + mn;
            const float bv = bias[col];
            float* const oj = obase + col;
            #pragma unroll
            for (int v = 0; v < 8; ++v) {
                oj[(size_t)v * DOUT] = fmaxf(acc[v] + bv, 0.0f);   // base + imm offset stores
            }
        }
    }
}

extern "C" void kernel_launch(void* const* d_in, const int* in_sizes, int n_in,
                              void* d_out, int out_size, void* d_ws, size_t ws_size,
                              hipStream_t stream) {
    const float* node = (const float*)d_in[0];   // [N, 256]
    const float* nbr  = (const float*)d_in[1];   // [N, 32, 256]
    const float* W    = (const float*)d_in[2];   // [256, 512]
    const float* bias = (const float*)d_in[3];   // [256]
    float* out        = (float*)d_out;           // [N, 256]

    const int grid = (NROWS + BLOCK_ROWS - 1) / BLOCK_ROWS;  // 1563
    sage_fused_wmma_kernel<<<grid, 256, 0, stream>>>(node, nbr, W, bias, out);
}